// P2RCountLoss_33646773797363
// MI455X (gfx1250) — compile-verified
//
#include <hip/hip_runtime.h>
#include <math.h>

// Problem constants (match reference)
#define BB 32
#define HH 512
#define WW 512
#define NN 4096
#define HWW (HH * WW)          // 262144
#define COUNT_W_C 2.0f
#define SPATIAL_W_C 0.15f

// exp(-0.5), exp(-sqrt(2)/2) computed in double, rounded to f32 (matches numpy)
#define W_AXIS 0.60653065971263342f
#define W_DIAG 0.49306869139523984f

__device__ __constant__ float c_splat[9] = {
    W_DIAG, W_AXIS, W_DIAG,
    W_AXIS, 1.0f,   W_AXIS,
    W_DIAG, W_AXIS, W_DIAG
};

typedef __attribute__((ext_vector_type(2))) float v2f;
typedef __attribute__((ext_vector_type(8))) float v8f;

// ---------------------------------------------------------------------------
// Kernel 1: zero the dense target scratch buffer (8388608 floats = 32 MiB)
// ---------------------------------------------------------------------------
__global__ __launch_bounds__(256) void zero_kernel(float4* __restrict__ p, int n4) {
    int stride = gridDim.x * blockDim.x;
    for (int i = blockIdx.x * blockDim.x + threadIdx.x; i < n4; i += stride) {
        p[i] = make_float4(0.f, 0.f, 0.f, 0.f);
    }
}

// ---------------------------------------------------------------------------
// Kernel 2: streaming reduction of pred -> per-block partial (sum, sum^2)
// via V_WMMA_F32_16X16X4_F32 with an all-ones B matrix.
// grid = 32 batches * 16 slices = 512 blocks, 256 threads (8 waves).
// Each block covers 16384 floats of one batch.
// ---------------------------------------------------------------------------
__global__ __launch_bounds__(256) void reduce_pred_kernel(
        const float* __restrict__ pred,
        float* __restrict__ psum, float* __restrict__ psum2) {
    const int b     = blockIdx.x >> 4;
    const int slice = blockIdx.x & 15;
    const float4* base = (const float4*)(pred + (size_t)b * HWW + (size_t)slice * 16384);

    v8f acc  = {};
    v8f acc2 = {};
    v2f ones; ones[0] = 1.0f; ones[1] = 1.0f;

    // 16 iterations * 256 threads * float4 = 16384 floats
    #pragma unroll 4
    for (int it = 0; it < 16; ++it) {
        const float4 v = base[it * 256 + threadIdx.x];
        v2f a1; a1[0] = v.x;        a1[1] = v.y;
        v2f a2; a2[0] = v.z;        a2[1] = v.w;
        v2f q1; q1[0] = v.x * v.x;  q1[1] = v.y * v.y;
        v2f q2; q2[0] = v.z * v.z;  q2[1] = v.w * v.w;
        // D += A x ones : every D element accumulates a row-sum of A.
        acc  = __builtin_amdgcn_wmma_f32_16x16x4_f32(false, a1, false, ones, (short)0, acc,  false, false);
        acc  = __builtin_amdgcn_wmma_f32_16x16x4_f32(false, a2, false, ones, (short)0, acc,  false, false);
        acc2 = __builtin_amdgcn_wmma_f32_16x16x4_f32(false, q1, false, ones, (short)0, acc2, false, false);
        acc2 = __builtin_amdgcn_wmma_f32_16x16x4_f32(false, q2, false, ones, (short)0, acc2, false, false);
    }

    // Sum of all 16x16 accumulator elements = 16 * (sum of all A elements fed).
    float ls = 0.f, ls2 = 0.f;
    #pragma unroll
    for (int i = 0; i < 8; ++i) { ls += acc[i]; ls2 += acc2[i]; }

    __shared__ float sh1[256];
    __shared__ float sh2[256];
    sh1[threadIdx.x] = ls;
    sh2[threadIdx.x] = ls2;
    __syncthreads();
    for (int s = 128; s > 0; s >>= 1) {
        if (threadIdx.x < s) {
            sh1[threadIdx.x] += sh1[threadIdx.x + s];
            sh2[threadIdx.x] += sh2[threadIdx.x + s];
        }
        __syncthreads();
    }
    if (threadIdx.x == 0) {
        psum [blockIdx.x] = sh1[0] * (1.0f / 16.0f);
        psum2[blockIdx.x] = sh2[0] * (1.0f / 16.0f);
    }
}

// ---------------------------------------------------------------------------
// Kernel 3: splat points. For each of the 9 neighbors:
//   T   += w                      (target mass)
//   PT  += pred[pix] * w          (cross term, distributes over entries)
//   T2  += 2*old*w + w*w          (telescopes to final per-pixel tw^2)
// grid = B*N/256 = 512 blocks; each block lies fully inside one batch.
// ---------------------------------------------------------------------------
__global__ __launch_bounds__(256) void scatter_kernel(
        const float* __restrict__ pred, const int* __restrict__ pts,
        float* __restrict__ tbuf,
        float* __restrict__ pT, float* __restrict__ pT2, float* __restrict__ pPT) {
    const int t = blockIdx.x * 256 + threadIdx.x;   // 0 .. B*N-1
    const int b = t >> 12;                          // / 4096
    int x = pts[2 * t + 0];
    int y = pts[2 * t + 1];
    x = min(max(x, 0), WW - 1);
    y = min(max(y, 0), HH - 1);

    float*       tb = tbuf + (size_t)b * HWW;
    const float* pb = pred + (size_t)b * HWW;

    float lt = 0.f, lt2 = 0.f, lpt = 0.f;
    #pragma unroll
    for (int k = 0; k < 9; ++k) {
        const int dy = k / 3 - 1;
        const int dx = k % 3 - 1;
        const int ny = y + dy;
        const int nx = x + dx;
        if (ny >= 0 && ny < HH && nx >= 0 && nx < WW) {
            const float w   = c_splat[k];
            const int   idx = ny * WW + nx;
            const float old = atomicAdd(&tb[idx], w);
            lt  += w;
            lt2 += w * (2.0f * old + w);
            lpt += pb[idx] * w;
        }
    }

    __shared__ float s0[256];
    __shared__ float s1[256];
    __shared__ float s2[256];
    s0[threadIdx.x] = lt;
    s1[threadIdx.x] = lt2;
    s2[threadIdx.x] = lpt;
    __syncthreads();
    for (int s = 128; s > 0; s >>= 1) {
        if (threadIdx.x < s) {
            s0[threadIdx.x] += s0[threadIdx.x + s];
            s1[threadIdx.x] += s1[threadIdx.x + s];
            s2[threadIdx.x] += s2[threadIdx.x + s];
        }
        __syncthreads();
    }
    if (threadIdx.x == 0) {
        pT [blockIdx.x] = s0[0];
        pT2[blockIdx.x] = s1[0];
        pPT[blockIdx.x] = s2[0];
    }
}

// ---------------------------------------------------------------------------
// Kernel 4: finalize. One block of 32 threads, thread b handles batch b.
//   spatial_b = (1/HW) * [ S2/S^2 - 2*PT/(S*T) + T2/T^2 ]
//   out = COUNT_W*sum|S/ca - N|/B + SPATIAL_W*sum(spatial_b)/B
// ---------------------------------------------------------------------------
__global__ __launch_bounds__(32) void finalize_kernel(
        const float* __restrict__ psum, const float* __restrict__ psum2,
        const float* __restrict__ pT,  const float* __restrict__ pT2,
        const float* __restrict__ pPT, const float* __restrict__ cell_area_ptr,
        float* __restrict__ out) {
    const int b = threadIdx.x;   // 0..31
    float s = 0.f, s2 = 0.f, T = 0.f, T2 = 0.f, PT = 0.f;
    #pragma unroll
    for (int j = 0; j < 16; ++j) {
        const int o = b * 16 + j;
        s  += psum [o];
        s2 += psum2[o];
        T  += pT  [o];
        T2 += pT2 [o];
        PT += pPT [o];
    }
    const float ca = *cell_area_ptr;
    const float S  = s + 1e-8f;
    const float cterm = fabsf(s / ca - (float)NN);
    const float sp = s2 / (S * S) - 2.0f * PT / (S * T) + T2 / (T * T);

    __shared__ float sc[32];
    __shared__ float ss[32];
    sc[b] = cterm;
    ss[b] = sp;
    __syncthreads();
    if (b == 0) {
        float cl = 0.f, sl = 0.f;
        #pragma unroll
        for (int k = 0; k < 32; ++k) { cl += sc[k]; sl += ss[k]; }
        out[0] = COUNT_W_C * (cl / (float)BB)
               + SPATIAL_W_C * ((sl / (float)HWW) / (float)BB);
    }
}

// ---------------------------------------------------------------------------
// Host launcher
// ---------------------------------------------------------------------------
extern "C" void kernel_launch(void* const* d_in, const int* in_sizes, int n_in,
                              void* d_out, int out_size, void* d_ws, size_t ws_size,
                              hipStream_t stream) {
    const float* pred = (const float*)d_in[0];   // (B,1,H,W) f32
    const int*   pts  = (const int*)  d_in[1];   // (B,N,2)   i32
    const float* ca   = (const float*)d_in[2];   // scalar    f32
    float*       out  = (float*)d_out;

    // Workspace layout (floats):
    //   [0, B*HW)              dense target accumulator (32 MiB)
    //   then 5 partial arrays of 512 floats each
    float* ws    = (float*)d_ws;
    float* tbuf  = ws;
    float* psum  = ws + (size_t)BB * HWW;
    float* psum2 = psum  + 512;
    float* pT    = psum2 + 512;
    float* pT2   = pT    + 512;
    float* pPT   = pT2   + 512;

    const int n4 = (BB * HWW) / 4;   // 2097152 float4
    zero_kernel<<<2048, 256, 0, stream>>>((float4*)tbuf, n4);
    reduce_pred_kernel<<<BB * 16, 256, 0, stream>>>(pred, psum, psum2);
    scatter_kernel<<<(BB * NN) / 256, 256, 0, stream>>>(pred, pts, tbuf, pT, pT2, pPT);
    finalize_kernel<<<1, 32, 0, stream>>>(psum, psum2, pT, pT2, pPT, ca, out);
}